// BilateralKernelLoss_87986700026273
// MI455X (gfx1250) — compile-verified
//
#include <hip/hip_runtime.h>
#include <math.h>

// ---------------------------------------------------------------------------
// BilateralKernelLoss for MI455X (gfx1250, wave32).
// Transcendental-bound stencil: LDS-tiled 5x5 bilateral filter on both
// tensors, fused |diff| + hierarchical reduction. Wave-level reductions use
// v_wmma_f32_16x16x4_f32 (A = lane partials, B = ones) -> row-sum broadcast.
// This round: raw v_exp_f32 via __builtin_amdgcn_exp2f (exponent arg is
// bounded in [-11,0], so OCML's subnormal fixup around exp2f is dead weight).
// ---------------------------------------------------------------------------

typedef float v2f __attribute__((ext_vector_type(2)));
typedef float v8f __attribute__((ext_vector_type(8)));

#define TS    16            // output tile edge
#define PADR  2             // halo = KS/2
#define TSH   (TS + 2*PADR) // 20
#define HW    512

#define LOG2E 1.44269504088896340736f
// w = exp(-(p-c)^2/(2*0.1) - (dx^2+dy^2)/(2*1.5))
//   = exp2( SOFF2 - (5*log2e)*(p-c)^2 ),  SOFF2 = -(dx^2+dy^2)/3 * log2e
#define SOFF2(dy, dx) (-(float)((((dx)-2)*((dx)-2)) + (((dy)-2)*((dy)-2))) * (LOG2E / 3.0f))
#define RANGE_C (5.0f * LOG2E)

// jnp.pad mode='reflect' (no edge duplication): -1 -> 1, H -> H-2
__device__ __forceinline__ int reflect_idx(int i, int n) {
    if (i < 0)  i = -i;
    if (i >= n) i = 2 * n - 2 - i;
    return i;
}

// ds_swizzle group-of-32 XOR pattern: offset = (xor_mask<<10) | 0x1f
__device__ __forceinline__ float swz_xor16(float x) {
    return __int_as_float(__builtin_amdgcn_ds_swizzle(__float_as_int(x), 0x401F));
}

// Full wave32 sum via one WMMA: A(16x4) v0 = per-lane value, v1 = 0; B = ones.
// D[m][n] = s[m] + s[m+16] broadcast into D's 8 VGPRs; per-lane sum of the 8
// regs gives the two half-wave totals, folded with a SWAPX16 swizzle.
// Requires EXEC all-ones (called from fully-active 256-thread blocks).
__device__ __forceinline__ float wave_sum_wmma(float s) {
    v2f a; a.x = s;    a.y = 0.0f;
    v2f b; b.x = 1.0f; b.y = 1.0f;
    v8f c = {};
    v8f d = __builtin_amdgcn_wmma_f32_16x16x4_f32(
        /*neg_a=*/false, a, /*neg_b=*/false, b,
        /*c_mod=*/(short)0, c, /*reuse_a=*/false, /*reuse_b=*/false);
    float r = d[0] + d[1] + d[2] + d[3] + d[4] + d[5] + d[6] + d[7];
    r += swz_xor16(r);
    return r;
}

__device__ __forceinline__ float wave_min(float m) {
    m = fminf(m, __int_as_float(__builtin_amdgcn_ds_swizzle(__float_as_int(m), 0x401F)));
    m = fminf(m, __int_as_float(__builtin_amdgcn_ds_swizzle(__float_as_int(m), 0x201F)));
    m = fminf(m, __int_as_float(__builtin_amdgcn_ds_swizzle(__float_as_int(m), 0x101F)));
    m = fminf(m, __int_as_float(__builtin_amdgcn_ds_swizzle(__float_as_int(m), 0x081F)));
    m = fminf(m, __int_as_float(__builtin_amdgcn_ds_swizzle(__float_as_int(m), 0x041F)));
    return m;
}

// ---- K1: per-block partial minimums of both tensors -----------------------
__global__ void k_min_partial(const float* __restrict__ A,
                              const float* __restrict__ B,
                              int n, float* __restrict__ partials) {
    const int half  = gridDim.x >> 1;            // 128 blocks per tensor
    const bool isA  = (int)blockIdx.x < half;
    const float* src = isA ? A : B;
    const int b      = isA ? blockIdx.x : (blockIdx.x - half);

    float m = 3.4e38f;
    for (int i = b * blockDim.x + threadIdx.x; i < n; i += half * blockDim.x)
        m = fminf(m, src[i]);

    m = wave_min(m);
    __shared__ float red[8];
    const int lane = threadIdx.x & 31, wid = threadIdx.x >> 5;
    if (lane == 0) red[wid] = m;
    __syncthreads();
    if (threadIdx.x == 0) {
        float t = red[0];
        #pragma unroll
        for (int i = 1; i < 8; ++i) t = fminf(t, red[i]);
        partials[blockIdx.x] = t;
    }
}

// ---- K2: finalize mins -> per-tensor (scale, bias) ------------------------
// img01 = min<0 ? (x+1)/2 : x  ==  x*scale + bias
__global__ void k_min_finalize(const float* __restrict__ partials,
                               float* __restrict__ sb) {
    __shared__ float red[256];
    const int tid = threadIdx.x;
    red[tid] = partials[tid];
    __syncthreads();
    #pragma unroll
    for (int s = 64; s >= 1; s >>= 1) {   // reduce each contiguous half of 128
        if ((tid & 127) < s) red[tid] = fminf(red[tid], red[tid + s]);
        __syncthreads();
    }
    if (tid == 0) {
        const float mA = red[0], mB = red[128];
        sb[0] = (mA < 0.0f) ? 0.5f : 1.0f;
        sb[1] = (mA < 0.0f) ? 0.5f : 0.0f;
        sb[2] = (mB < 0.0f) ? 0.5f : 1.0f;
        sb[3] = (mB < 0.0f) ? 0.5f : 0.0f;
    }
}

// ---- K3: tiled bilateral filter on both tensors + |diff| block sums -------
__global__ void __launch_bounds__(256)
k_bilateral(const float* __restrict__ A, const float* __restrict__ B,
            const float* __restrict__ sb, float* __restrict__ blockSums) {
    __shared__ float tA[TSH][TSH];
    __shared__ float tB[TSH][TSH];

    const int tx = threadIdx.x, ty = threadIdx.y;
    const int tid = ty * TS + tx;
    const int x0 = blockIdx.x * TS, y0 = blockIdx.y * TS;
    const long long planeBase = (long long)blockIdx.z * (HW * HW);

    const float sA = sb[0], bA = sb[1], sBc = sb[2], bB = sb[3];

    // Stage 20x20 halo tiles (reflect padding), pre-remapped to [0,1] domain.
    for (int idx = tid; idx < TSH * TSH; idx += TS * TS) {
        const int ly = idx / TSH, lx = idx % TSH;
        const int gy = reflect_idx(y0 + ly - PADR, HW);
        const int gx = reflect_idx(x0 + lx - PADR, HW);
        const long long g = planeBase + (long long)gy * HW + gx;
        tA[ly][lx] = A[g] * sA  + bA;
        tB[ly][lx] = B[g] * sBc + bB;
    }
    __syncthreads();

    const float cA = tA[ty + PADR][tx + PADR];
    const float cB = tB[ty + PADR][tx + PADR];
    float wsA = 0.0f, wpA = 0.0f, wsB = 0.0f, wpB = 0.0f;

    #pragma unroll
    for (int k = 0; k < 25; ++k) {
        const int dy = k / 5, dx = k % 5;
        const float off2 = SOFF2(dy, dx);      // compile-time literal
        const float p  = tA[ty + dy][tx + dx];
        const float d  = p - cA;
        // raw v_exp_f32: arg in [-11, 0], no range fixup needed
        const float w  = __builtin_amdgcn_exp2f(fmaf(d * d, -RANGE_C, off2));
        wsA += w; wpA = fmaf(w, p, wpA);
        const float q  = tB[ty + dy][tx + dx];
        const float e  = q - cB;
        const float v  = __builtin_amdgcn_exp2f(fmaf(e * e, -RANGE_C, off2));
        wsB += v; wpB = fmaf(v, q, wpB);
    }

    // v_rcp_f32 instead of IEEE-correct division chain (trans op, co-executes)
    const float fA = wpA * __builtin_amdgcn_rcpf(wsA + 1e-8f);
    const float fB = wpB * __builtin_amdgcn_rcpf(wsB + 1e-8f);
    float diff = fabsf(fA - fB);

    // Block reduction: WMMA wave sums -> 8 partials -> serial fixed-order sum.
    diff = wave_sum_wmma(diff);
    __shared__ float wsum[8];
    const int lane = tid & 31, wid = tid >> 5;
    if (lane == 0) wsum[wid] = diff;
    __syncthreads();
    if (tid == 0) {
        float t = 0.0f;
        #pragma unroll
        for (int i = 0; i < 8; ++i) t += wsum[i];
        const int bi = (blockIdx.z * gridDim.y + blockIdx.y) * gridDim.x + blockIdx.x;
        blockSums[bi] = t;
    }
}

// ---- K4: deterministic final sum -> mean ----------------------------------
__global__ void k_final_sum(const float* __restrict__ blockSums, int nb,
                            float* __restrict__ out, float invN) {
    float s = 0.0f;
    for (int i = threadIdx.x; i < nb; i += blockDim.x)  // fixed order
        s += blockSums[i];
    s = wave_sum_wmma(s);
    __shared__ float wsum[8];
    const int lane = threadIdx.x & 31, wid = threadIdx.x >> 5;
    if (lane == 0) wsum[wid] = s;
    __syncthreads();
    if (threadIdx.x == 0) {
        float t = 0.0f;
        #pragma unroll
        for (int i = 0; i < 8; ++i) t += wsum[i];
        out[0] = t * invN;
    }
}

// ---------------------------------------------------------------------------
extern "C" void kernel_launch(void* const* d_in, const int* in_sizes, int n_in,
                              void* d_out, int out_size, void* d_ws, size_t ws_size,
                              hipStream_t stream) {
    const float* A = (const float*)d_in[0];
    const float* B = (const float*)d_in[1];
    const int n = in_sizes[0];                 // 2*3*512*512 = 1,572,864
    const int planes = n / (HW * HW);          // 6
    const int tiles = HW / TS;                 // 32

    float* ws        = (float*)d_ws;
    float* partials  = ws;                     // [256]
    float* sb        = ws + 256;               // [4]
    float* blockSums = ws + 512;               // [tiles*tiles*planes]

    k_min_partial<<<256, 256, 0, stream>>>(A, B, n, partials);
    k_min_finalize<<<1, 256, 0, stream>>>(partials, sb);

    dim3 grid(tiles, tiles, planes), blk(TS, TS);
    k_bilateral<<<grid, blk, 0, stream>>>(A, B, sb, blockSums);

    const int nb = tiles * tiles * planes;     // 6144
    k_final_sum<<<1, 256, 0, stream>>>(blockSums, nb, (float*)d_out, 1.0f / (float)n);
}